// ObjectDetector_84387517432265
// MI455X (gfx1250) — compile-verified
//
#include <hip/hip_runtime.h>
#include <cstdint>

// ---------------------------------------------------------------------------
// YOLO-style post-processing (score -> histogram top-k -> greedy NMS -> top300)
// for MI455X / gfx1250.  Bandwidth-bound: one 261MB pass over pred dominates
// (~11us at 23.3TB/s); no dense GEMM exists in this op, so no WMMA.  CDNA5
// paths exercised instead:
//   * TDM: tensor_load_to_lds / tensor_store_from_lds + s_wait_tensorcnt
//     for the 32KB record block and the 7.2KB output block
//   * async copy: global_load_async_to_lds_b128 + s_wait_asynccnt (histogram)
//   * global_prefetch_b8 in the streaming score pass
// ---------------------------------------------------------------------------

#define BB      16
#define NN      102000
#define NCLS    35
#define ROWF    40          // floats per anchor row
#define NBINS   2048
#define PRETOPK 1024
#define MAXDET  300
#define CONF_T  0.25f
#define IOU_T   0.45f

// ---- workspace layout (bytes) ----
#define WS_SCORES 0                                   // BB*NN floats
#define WS_HIST   (WS_SCORES + BB*NN*4)               // BB*NBINS ints
#define WS_CUT    (WS_HIST + BB*NBINS*4)              // BB ints
#define WS_CNT    (WS_CUT + 64)                       // BB ints
#define WS_CAND   (WS_CNT + 64)                       // BB*PRETOPK*8 floats

#if defined(__has_builtin)
#  if __has_builtin(__builtin_amdgcn_tensor_load_to_lds)
#    define HAVE_TDM 1
#  endif
#endif
#ifndef HAVE_TDM
#  define HAVE_TDM 0
#endif

typedef unsigned int u32x4 __attribute__((ext_vector_type(4)));
typedef int          i32x8 __attribute__((ext_vector_type(8)));
typedef int          i32x4 __attribute__((ext_vector_type(4)));

#if HAVE_TDM
// Build a D# for a 1-D contiguous tile of `nelem` 4-byte elements.
// group0: [1:0]=count=1, [63:32]=lds_addr, [120:64]=global_addr, [127:126]=type(2)
// group1: [17:16]=data_size(2 -> 4B), [79:48]=tensor_dim0, [111:80]=tensor_dim1(1),
//         [127:112]=tile_dim0, [143:128]=tile_dim1(1), [207:160]=tensor_dim0_stride
__device__ __forceinline__ void tdm_desc_1d(unsigned lds_addr, unsigned long long ga,
                                            unsigned nelem, u32x4& g0, i32x8& g1) {
    g0.x = 1u;
    g0.y = lds_addr;
    g0.z = (unsigned)(ga & 0xffffffffull);
    g0.w = (unsigned)((ga >> 32) & 0x01ffffffull) | (2u << 30);
    g1[0] = (int)(2u << 16);                                   // data_size = 4B
    g1[1] = (int)((nelem & 0xffffu) << 16);                    // tensor_dim0 lo16
    g1[2] = (int)(((nelem >> 16) & 0xffffu) | (1u << 16));     // dim0 hi16 | dim1=1
    g1[3] = (int)((nelem & 0xffffu) << 16);                    // tile_dim0 (nelem<65536)
    g1[4] = 1;                                                 // tile_dim1 = 1
    g1[5] = (int)nelem;                                        // dim0 stride lo32
    g1[6] = 0;
    g1[7] = 0;
}
#endif

// ---------------------------------------------------------------------------
__global__ void k_zero(int* __restrict__ hist, int* __restrict__ cnt) {
    int i = blockIdx.x * blockDim.x + threadIdx.x;
    if (i < BB * NBINS) hist[i] = 0;
    if (i < BB)         cnt[i]  = 0;
}

// ---------------------------------------------------------------------------
// Pass 1: score every anchor, build per-batch score histogram.
__global__ void k_score(const float* __restrict__ pred,
                        float* __restrict__ scores,
                        int*   __restrict__ hist) {
    int idx = blockIdx.x * blockDim.x + threadIdx.x;
    if (idx >= BB * NN) return;
    int b = idx / NN;

    const float* row = pred + (size_t)idx * ROWF;
    __builtin_prefetch(row + 16 * ROWF, 0, 3);        // global_prefetch_b8

    const float4* r4 = (const float4*)row;
    float4 r1 = r4[1];                                 // obj, c0, c1, c2
    float obj = r1.x;
    float m = fmaxf(r1.y, fmaxf(r1.z, r1.w));
#pragma unroll
    for (int c = 2; c < 10; ++c) {
        float4 rc = r4[c];
        m = fmaxf(m, fmaxf(fmaxf(rc.x, rc.y), fmaxf(rc.z, rc.w)));
    }
    float conf = obj * m;
    bool cand = (obj > CONF_T) && (conf > CONF_T);
    float s = cand ? conf : 0.0f;
    scores[idx] = s;
    if (s > 0.0f) {
        int bin = (int)(s * (float)NBINS);
        bin = bin > NBINS - 1 ? NBINS - 1 : bin;
        atomicAdd(&hist[b * NBINS + bin], 1);
    }
}

// ---------------------------------------------------------------------------
// Pass 2: one workgroup per batch.  Async-stage the 8KB histogram into LDS
// (CDNA5 global_load_async_to_lds path), then scan top-down for the bin
// cutoff holding the PRETOPK-th best score.
__global__ void k_cut(const int* __restrict__ hist, int* __restrict__ cut) {
    __shared__ __align__(16) int sh[NBINS];
    const int b = blockIdx.x;
    const int t = threadIdx.x;                         // 256 threads, 8 ints each
    {
        unsigned lds = (unsigned)(uintptr_t)(&sh[t * 8]);
        unsigned long long ga =
            (unsigned long long)(uintptr_t)(hist + b * NBINS + t * 8);
        asm volatile("global_load_async_to_lds_b128 %0, %1, off\n\t"
                     "global_load_async_to_lds_b128 %0, %1, off offset:16"
                     :: "v"(lds), "v"(ga) : "memory");
        asm volatile("s_wait_asynccnt 0" ::: "memory");
    }
    __syncthreads();
    if (t == 0) {
        int cum = 0, c = 0;
        for (int bin = NBINS - 1; bin >= 0; --bin) {
            cum += sh[bin];
            if (cum >= PRETOPK) { c = bin; break; }
        }
        cut[b] = c;
    }
}

// ---------------------------------------------------------------------------
// Pass 3: collect accepted anchors into compact 32B records.
__global__ void k_collect(const float* __restrict__ pred,
                          const float* __restrict__ scores,
                          const int*   __restrict__ cut,
                          int*   __restrict__ cnt,
                          float* __restrict__ cand) {
    int idx = blockIdx.x * blockDim.x + threadIdx.x;
    if (idx >= BB * NN) return;
    int b = idx / NN;
    int n = idx - b * NN;

    float s = scores[idx];
    if (s <= 0.0f) return;
    int bin = (int)(s * (float)NBINS);
    bin = bin > NBINS - 1 ? NBINS - 1 : bin;
    if (bin < cut[b]) return;

    int slot = atomicAdd(&cnt[b], 1);
    if (slot >= PRETOPK) return;

    const float4* r4 = (const float4*)(pred + (size_t)idx * ROWF);
    float4 r0 = r4[0];                                 // x, y, w, h
    float4 r1 = r4[1];                                 // obj, c0, c1, c2
    float m = r1.y; int mi = 0;
    if (r1.z > m) { m = r1.z; mi = 1; }
    if (r1.w > m) { m = r1.w; mi = 2; }
#pragma unroll
    for (int c = 2; c < 10; ++c) {
        float4 rc = r4[c];
        int base = 4 * c - 5;                          // class idx of rc.x
        if (rc.x > m) { m = rc.x; mi = base; }
        if (rc.y > m) { m = rc.y; mi = base + 1; }
        if (rc.z > m) { m = rc.z; mi = base + 2; }
        if (rc.w > m) { m = rc.w; mi = base + 3; }
    }
    float hw = r0.z * 0.5f, hh = r0.w * 0.5f;
    float4* o = (float4*)(cand + ((size_t)b * PRETOPK + slot) * 8);
    o[0] = make_float4(r0.x - hw, r0.y - hh, r0.x + hw, r0.y + hh);
    o[1] = make_float4(s, (float)mi, (float)n, 0.0f);
}

// ---------------------------------------------------------------------------
// Pass 4: per-batch sort (bitonic, LDS) + greedy NMS + top-300 emit.
// TDM moves the 32KB record block global->LDS and the 7.2KB result block
// LDS->global; one descriptor each, issued by wave 0.
__global__ __launch_bounds__(PRETOPK) void k_nms(const float* __restrict__ cand,
                                                 const int*   __restrict__ cnt,
                                                 float* __restrict__ out) {
    __shared__ __align__(16) float recf[PRETOPK * 8];          // 32KB (reused)
    __shared__ __align__(16) unsigned long long keys[PRETOPK]; // 8KB  (reused)
    const int b = blockIdx.x;
    const int t = threadIdx.x;
    const int cv = min(cnt[b], PRETOPK);

    // --- stage records: global -> LDS --------------------------------------
#if HAVE_TDM
    if (t < 32) {                                      // wave 0 only (EXEC skip)
        u32x4 g0; i32x8 g1;
        i32x4 gz  = {0, 0, 0, 0};
        i32x8 gz8 = {0, 0, 0, 0, 0, 0, 0, 0};
        tdm_desc_1d((unsigned)(uintptr_t)recf,
                    (unsigned long long)(uintptr_t)(cand + (size_t)b * PRETOPK * 8),
                    PRETOPK * 8, g0, g1);
        __builtin_amdgcn_tensor_load_to_lds(g0, g1, gz, gz, gz8, 0);
        __builtin_amdgcn_s_wait_tensorcnt(0);
    }
#else
    if (t < cv) {
        unsigned lds = (unsigned)(uintptr_t)(&recf[t * 8]);
        unsigned long long ga =
            (unsigned long long)(uintptr_t)(cand + ((size_t)b * PRETOPK + t) * 8);
        asm volatile("global_load_async_to_lds_b128 %0, %1, off\n\t"
                     "global_load_async_to_lds_b128 %0, %1, off offset:16"
                     :: "v"(lds), "v"(ga) : "memory");
    }
    asm volatile("s_wait_asynccnt 0" ::: "memory");
#endif
    __syncthreads();
    if (t >= cv) {                                     // zero-pad stale tail
#pragma unroll
        for (int k = 0; k < 8; ++k) recf[t * 8 + k] = 0.0f;
    }
    __syncthreads();

    // --- sort keys: (score_bits << 32) | slot, descending ------------------
    {
        float s = recf[t * 8 + 4];
        keys[t] = ((unsigned long long)__float_as_uint(s) << 32) | (unsigned)t;
    }
    __syncthreads();
    for (unsigned k = 2; k <= PRETOPK; k <<= 1) {
        for (unsigned j = k >> 1; j > 0; j >>= 1) {
            unsigned ixj = (unsigned)t ^ j;
            if (ixj > (unsigned)t) {
                unsigned long long a = keys[t], c = keys[ixj];
                bool up = ((t & k) == 0);
                bool sw = up ? (a < c) : (a > c);               // descending
                if (sw) { keys[t] = c; keys[ixj] = a; }
            }
            __syncthreads();
        }
    }

    // --- gather sorted record into registers -------------------------------
    unsigned slot = (unsigned)(keys[t] & 0xffffffffu);
    float x1 = recf[slot * 8 + 0], y1 = recf[slot * 8 + 1];
    float x2 = recf[slot * 8 + 2], y2 = recf[slot * 8 + 3];
    float sc = recf[slot * 8 + 4], cl = recf[slot * 8 + 5];
    __syncthreads();                                           // before reuse

    // --- reuse recf region: boxes + keep + pos -----------------------------
    float4* sbox = (float4*)recf;                              // [1024]
    int*    keep = (int*)(recf + PRETOPK * 4);                 // [1024]
    int*    pos  = (int*)(recf + PRETOPK * 5);                 // [1024]
    sbox[t] = make_float4(x1, y1, x2, y2);
    __syncthreads();

    // --- greedy NMS, exact reference-scan semantics ------------------------
    float area = fmaxf(x2 - x1, 0.f) * fmaxf(y2 - y1, 0.f);
    bool sup = false;
    bool valid = sc > 0.0f;
    for (int i = 0; i < PRETOPK; ++i) {
        if (t == i) keep[i] = (valid && !sup) ? 1 : 0;
        __syncthreads();
        if (keep[i] && t > i) {
            float4 bi = sbox[i];
            float ax = fmaxf(bi.x, x1), ay = fmaxf(bi.y, y1);
            float bx = fminf(bi.z, x2), by = fminf(bi.w, y2);
            float iw = fmaxf(bx - ax, 0.f), ih = fmaxf(by - ay, 0.f);
            float inter = iw * ih;
            float ai = fmaxf(bi.z - bi.x, 0.f) * fmaxf(bi.w - bi.y, 0.f);
            float iou = inter / (ai + area - inter + 1e-7f);
            if (iou > IOU_T) sup = true;
        }
        __syncthreads();
    }

    // --- output ordering: kept (score order) first, then zero-score fill ---
    if (t == 0) {
        int kc = 0;
        for (int i = 0; i < PRETOPK; ++i) if (keep[i])  pos[i] = kc++;
        int z = kc;
        for (int i = 0; i < PRETOPK; ++i) if (!keep[i]) pos[i] = z++;
    }
    __syncthreads();
    int p = pos[t];
    bool kp = keep[t] != 0;

    // stage the 300x6 result block in LDS (reuse key region), then TDM-store
    float* outSh = (float*)keys;                               // 7200B <= 8KB
    if (p < MAXDET) {
        outSh[p * 6 + 0] = x1; outSh[p * 6 + 1] = y1;
        outSh[p * 6 + 2] = x2; outSh[p * 6 + 3] = y2;
        outSh[p * 6 + 4] = kp ? sc : 0.0f;
        outSh[p * 6 + 5] = cl;
    }
    __syncthreads();
#if HAVE_TDM
    if (t < 32) {                                      // wave 0 only
        u32x4 g0; i32x8 g1;
        i32x4 gz  = {0, 0, 0, 0};
        i32x8 gz8 = {0, 0, 0, 0, 0, 0, 0, 0};
        tdm_desc_1d((unsigned)(uintptr_t)outSh,
                    (unsigned long long)(uintptr_t)(out + (size_t)b * MAXDET * 6),
                    MAXDET * 6, g0, g1);
        __builtin_amdgcn_tensor_store_from_lds(g0, g1, gz, gz, gz8, 0);
        __builtin_amdgcn_s_wait_tensorcnt(0);
    }
#else
    if (t < MAXDET * 6 / 4) {                          // 450 float4 stores
        float4* src = (float4*)outSh;
        ((float4*)(out + (size_t)b * MAXDET * 6))[t] = src[t];
    }
#endif
}

// ---------------------------------------------------------------------------
extern "C" void kernel_launch(void* const* d_in, const int* in_sizes, int n_in,
                              void* d_out, int out_size, void* d_ws, size_t ws_size,
                              hipStream_t stream) {
    (void)in_sizes; (void)n_in; (void)out_size; (void)ws_size;
    const float* pred = (const float*)d_in[0];
    float* out = (float*)d_out;
    char*  ws  = (char*)d_ws;

    float* scores = (float*)(ws + WS_SCORES);
    int*   hist   = (int*)  (ws + WS_HIST);
    int*   cut    = (int*)  (ws + WS_CUT);
    int*   cnt    = (int*)  (ws + WS_CNT);
    float* cand   = (float*)(ws + WS_CAND);

    const int total = BB * NN;
    k_zero   <<<(BB * NBINS + 255) / 256, 256, 0, stream>>>(hist, cnt);
    k_score  <<<(total + 255) / 256, 256, 0, stream>>>(pred, scores, hist);
    k_cut    <<<BB, 256, 0, stream>>>(hist, cut);
    k_collect<<<(total + 255) / 256, 256, 0, stream>>>(pred, scores, cut, cnt, cand);
    k_nms    <<<BB, PRETOPK, 0, stream>>>(cand, cnt, out);
}